// EfficientDiscriminativeFrequencyDomainBasedFFN_60455959658563
// MI455X (gfx1250) — compile-verified
//
#include <hip/hip_runtime.h>
#include <hip/hip_bf16.h>

typedef float v2f __attribute__((ext_vector_type(2)));
typedef float v8f __attribute__((ext_vector_type(8)));

#define DIM 64
#define HID 32
#define IMG_H 512
#define IMG_W 512
#define NBATCH 4
#define TH 4             // tile rows (== FFT block P)
#define TW 64            // tile cols (16 FFT blocks)
#define HR 6             // halo rows
#define HC 66            // halo cols (1 + 64 + 1)
#define SROW 72          // padded LDS row stride: [0..2]=pad, 3=left halo, 4..67=interior, 68=right halo, [69..71]=pad
#define CST (HR * SROW)  // 432 floats per channel
#define NPT1 27          // 432/16 N-tiles for conv1 (exact, no tail)
#define NT2 (TH * TW)    // 256 interior positions
#define NTHREADS 512

__global__ __launch_bounds__(NTHREADS)
void edfdffn_fused(const float* __restrict__ x,
                   const float* __restrict__ w_in1,
                   const float* __restrict__ w_dw,
                   const float* __restrict__ w_in2,
                   const float* __restrict__ w_freq,
                   float* __restrict__ out)
{
    __shared__ __attribute__((aligned(16))) float xs[DIM * CST];  // x tile (padded halo layout)
    __shared__ __attribute__((aligned(16))) float ys[DIM * CST];  // conv1 out; reused as zs[DIM*NT2]
    __shared__ __attribute__((aligned(16))) float gs[HID * NT2];  // gated hidden
    __shared__ float w1s[DIM * DIM];
    __shared__ float w2s[DIM * HID];
    __shared__ float wdws[DIM * 9];
    __shared__ float wfs[DIM * 12];

    const int tid  = threadIdx.x;
    const int wave = tid >> 5;
    const int lane = tid & 31;
    const int half = lane >> 4;   // 0: lanes 0-15, 1: lanes 16-31
    const int ln   = lane & 15;

    const int w0 = blockIdx.x * TW;
    const int h0 = blockIdx.y * TH;
    const int b  = blockIdx.z;

    const float* xb = x + (size_t)b * DIM * IMG_H * IMG_W;

    // ---------------- Stage A: stage x tile into LDS ------------------
    const bool interior = (h0 > 0) && (h0 + TH < IMG_H) && (w0 > 0) && (w0 + TW < IMG_W);
    if (interior) {
        // (1) zero 6 columns per row: pad 0..2 and cols 66..68 (66..68 are also async destinations)
        for (int i = tid; i < DIM * HR * 6; i += NTHREADS) {
            int c = i / 36, rem = i - c * 36;
            int rr = rem / 6, j = rem - rr * 6;
            int col = (j < 3) ? j : (j + 63);           // 0,1,2,66,67,68
            xs[c * CST + rr * SROW + col] = 0.f;
        }
        // (2) async DMA global -> LDS: per (c,row): 16x b128 interior + 2x b32 halo
        const uint64_t gbase = (uint64_t)(uintptr_t)xb;
        for (int i = tid; i < DIM * HR * 18; i += NTHREADS) {
            int c = i / (HR * 18), rem = i - c * (HR * 18);
            int r = rem / 18, u = rem - r * 18;
            int gh = h0 - 1 + r;                         // always in range (interior WG)
            uint64_t rowg = gbase + (((uint64_t)c * IMG_H + (uint64_t)gh) * IMG_W) * 4ull;
            uint32_t rowl = (uint32_t)(uintptr_t)&xs[c * CST + r * SROW];
            if (u == 0) {                                // left halo (gw = w0-1) -> slot 3
                uint64_t ga = rowg + (uint64_t)(w0 - 1) * 4ull;
                uint32_t la = rowl + 3u * 4u;
                asm volatile("global_load_async_to_lds_b32 %0, %1, off"
                             :: "v"(la), "v"(ga) : "memory");
            } else if (u == 17) {                        // right halo (gw = w0+64) -> slot 68
                uint64_t ga = rowg + (uint64_t)(w0 + TW) * 4ull;
                uint32_t la = rowl + 68u * 4u;
                asm volatile("global_load_async_to_lds_b32 %0, %1, off"
                             :: "v"(la), "v"(ga) : "memory");
            } else {                                     // 16B-aligned interior chunk
                int t = u - 1;                           // 0..15, gw = w0 + 4t
                uint64_t ga = rowg + (uint64_t)(w0 + 4 * t) * 4ull;
                uint32_t la = rowl + (uint32_t)(4 + 4 * t) * 4u;
                asm volatile("global_load_async_to_lds_b128 %0, %1, off"
                             :: "v"(la), "v"(ga) : "memory");
            }
        }
    } else {
        // border WG: scalar path with zero fill outside the image
        for (int i = tid; i < DIM * CST; i += NTHREADS) {
            int c = i / CST, rem = i - c * CST;
            int r = rem / SROW, col = rem - r * SROW;
            float v = 0.f;
            int hcol = col - 3;                          // 0..65 maps to gw = w0-1+hcol
            if ((unsigned)hcol < HC) {
                int gh = h0 - 1 + r, gw = w0 - 1 + hcol;
                if ((unsigned)gh < IMG_H && (unsigned)gw < IMG_W)
                    v = xb[(size_t)c * IMG_H * IMG_W + (size_t)gh * IMG_W + gw];
            }
            xs[i] = v;
        }
    }
    for (int i = tid; i < DIM * DIM; i += NTHREADS) w1s[i] = w_in1[i];
    for (int i = tid; i < DIM * HID; i += NTHREADS) w2s[i] = w_in2[i];
    for (int i = tid; i < DIM * 9;   i += NTHREADS) wdws[i] = w_dw[i];
    for (int i = tid; i < DIM * 12;  i += NTHREADS) wfs[i]  = w_freq[i];
#if __has_builtin(__builtin_amdgcn_s_wait_asynccnt)
    __builtin_amdgcn_s_wait_asynccnt(0);
#else
    asm volatile("s_wait_asynccnt 0x0" ::: "memory");
#endif
    __syncthreads();

    // ------------- Stage B: conv1x1 #1 via V_WMMA_F32_16X16X4_F32 ------
    // ys[oc][p] = sum_ic w1[oc][ic] * xs[ic][p];  M=64, N=432(padded), K=64
    // Each wave owns one M-tile (mt = wave&3): hoist its 16 A-fragments.
    {
        const int mt = wave & 3, oc0 = mt * 16;
        v2f a[16];
        #pragma unroll
        for (int ks = 0; ks < 16; ++ks) {
            // A 16x4 (f32): v0 = K=0 (lanes0-15) / K=2 (lanes16-31); v1 = K=1 / K=3
            a[ks].x = w1s[(oc0 + ln) * DIM + 4 * ks + (half ? 2 : 0)];
            a[ks].y = w1s[(oc0 + ln) * DIM + 4 * ks + (half ? 3 : 1)];
        }
        for (int nt = wave >> 2; nt < NPT1; nt += 4) {
            int p0 = nt * 16, pb = p0 + ln;
            v8f acc = {0.f, 0.f, 0.f, 0.f, 0.f, 0.f, 0.f, 0.f};
            #pragma unroll
            for (int ks = 0; ks < 16; ++ks) {
                v2f bb;                                  // B 4x16: same K split, N = lane%16
                bb.x = xs[(4 * ks + (half ? 2 : 0)) * CST + pb];
                bb.y = xs[(4 * ks + (half ? 3 : 1)) * CST + pb];
                acc = __builtin_amdgcn_wmma_f32_16x16x4_f32(
                          false, a[ks], false, bb, (short)0, acc, false, false);
            }
            #pragma unroll
            for (int j = 0; j < 8; ++j)                  // C/D: VGPR j -> M = j (+8 hi lanes)
                ys[(oc0 + j + half * 8) * CST + pb] = acc[j];
        }
    }
    __syncthreads();

    // ------------- Stage C: depthwise 3x3 + exact-GELU gate ------------
    // 4 outputs per thread; window rows loaded once (6 values per row).
    for (int i = tid; i < HID * TH * 16; i += NTHREADS) {
        int hc = i >> 6;
        int rem = i & 63, r = rem >> 4, q = rem & 15;
        int cc0 = q * 4;
        int base = r * SROW + 3 + cc0;                   // window left col (stored coords)
        float y1[4] = {0.f,0.f,0.f,0.f}, y2[4] = {0.f,0.f,0.f,0.f};
        #pragma unroll
        for (int kh = 0; kh < 3; ++kh) {
            float a1[6], a2[6];
            #pragma unroll
            for (int j = 0; j < 6; ++j) {
                a1[j] = ys[hc * CST + base + kh * SROW + j];
                a2[j] = ys[(hc + HID) * CST + base + kh * SROW + j];
            }
            #pragma unroll
            for (int cc = 0; cc < 4; ++cc)
                #pragma unroll
                for (int kw = 0; kw < 3; ++kw) {
                    y1[cc] += wdws[hc * 9 + kh * 3 + kw]         * a1[cc + kw];
                    y2[cc] += wdws[(hc + HID) * 9 + kh * 3 + kw] * a2[cc + kw];
                }
        }
        #pragma unroll
        for (int cc = 0; cc < 4; ++cc) {
            float g = 0.5f * y1[cc] * (1.f + erff(y1[cc] * 0.70710678118654752f));
            gs[hc * NT2 + r * 64 + cc0 + cc] = g * y2[cc];
        }
    }
    __syncthreads();

    // ------------- Stage D: conv1x1 #2 via WMMA -> zs (aliases ys) -----
    float* zs = ys;
    {
        const int mt = wave & 3, oc0 = mt * 16;
        v2f a[8];
        #pragma unroll
        for (int ks = 0; ks < 8; ++ks) {
            a[ks].x = w2s[(oc0 + ln) * HID + 4 * ks + (half ? 2 : 0)];
            a[ks].y = w2s[(oc0 + ln) * HID + 4 * ks + (half ? 3 : 1)];
        }
        for (int nt = wave >> 2; nt < 16; nt += 4) {
            int p0 = nt * 16;
            v8f acc = {0.f, 0.f, 0.f, 0.f, 0.f, 0.f, 0.f, 0.f};
            #pragma unroll
            for (int ks = 0; ks < 8; ++ks) {
                v2f bb;
                bb.x = gs[(4 * ks + (half ? 2 : 0)) * NT2 + p0 + ln];
                bb.y = gs[(4 * ks + (half ? 3 : 1)) * NT2 + p0 + ln];
                acc = __builtin_amdgcn_wmma_f32_16x16x4_f32(
                          false, a[ks], false, bb, (short)0, acc, false, false);
            }
            #pragma unroll
            for (int j = 0; j < 8; ++j)
                zs[(oc0 + j + half * 8) * NT2 + p0 + ln] = acc[j];
        }
    }
    __syncthreads();

    // ------------- Stage E: 4x4-block spectral filter + shortcut -------
    for (int i = tid; i < DIM * 16; i += NTHREADS) {
        int c = i >> 4, bw = i & 15;
        float xv[4][4];
        #pragma unroll
        for (int r = 0; r < 4; ++r)
            #pragma unroll
            for (int cc = 0; cc < 4; ++cc)
                xv[r][cc] = zs[c * NT2 + r * 64 + bw * 4 + cc];
        float wfr[4][3];
        #pragma unroll
        for (int fh = 0; fh < 4; ++fh)
            #pragma unroll
            for (int fw = 0; fw < 3; ++fw)
                wfr[fh][fw] = wfs[c * 12 + fh * 3 + fw];

        // rfft along width per row: R0 real, R1 complex, R2 real
        float R0[4], R1re[4], R1im[4], R2[4];
        #pragma unroll
        for (int r = 0; r < 4; ++r) {
            float s02 = xv[r][0] + xv[r][2], d02 = xv[r][0] - xv[r][2];
            float s13 = xv[r][1] + xv[r][3], d13 = xv[r][1] - xv[r][3];
            R0[r] = s02 + s13;  R2[r] = s02 - s13;
            R1re[r] = d02;      R1im[r] = -d13;
        }
        // fft along height + multiply by w  => H[fh][fw]
        float Hre[4][3], Him[4][3];
        {   float A = R0[0] + R0[2], Bv = R0[0] - R0[2];
            float C = R0[1] + R0[3], D  = R0[1] - R0[3];
            Hre[0][0] = wfr[0][0] * (A + C); Him[0][0] = 0.f;
            Hre[1][0] = wfr[1][0] * Bv;      Him[1][0] = -wfr[1][0] * D;
            Hre[2][0] = wfr[2][0] * (A - C); Him[2][0] = 0.f;
            Hre[3][0] = wfr[3][0] * Bv;      Him[3][0] =  wfr[3][0] * D; }
        {   float A = R2[0] + R2[2], Bv = R2[0] - R2[2];
            float C = R2[1] + R2[3], D  = R2[1] - R2[3];
            Hre[0][2] = wfr[0][2] * (A + C); Him[0][2] = 0.f;
            Hre[1][2] = wfr[1][2] * Bv;      Him[1][2] = -wfr[1][2] * D;
            Hre[2][2] = wfr[2][2] * (A - C); Him[2][2] = 0.f;
            Hre[3][2] = wfr[3][2] * Bv;      Him[3][2] =  wfr[3][2] * D; }
        {   float Are = R1re[0] + R1re[2], Aim = R1im[0] + R1im[2];
            float Bre = R1re[0] - R1re[2], Bim = R1im[0] - R1im[2];
            float Cre = R1re[1] + R1re[3], Cim = R1im[1] + R1im[3];
            float Dre = R1re[1] - R1re[3], Dim = R1im[1] - R1im[3];
            Hre[0][1] = wfr[0][1] * (Are + Cre); Him[0][1] = wfr[0][1] * (Aim + Cim);
            Hre[1][1] = wfr[1][1] * (Bre + Dim); Him[1][1] = wfr[1][1] * (Bim - Dre);
            Hre[2][1] = wfr[2][1] * (Are - Cre); Him[2][1] = wfr[2][1] * (Aim - Cim);
            Hre[3][1] = wfr[3][1] * (Bre - Dim); Him[3][1] = wfr[3][1] * (Bim + Dre); }
        // ifft along height: T[r] = H0 + i^r*H1 + (-1)^r*H2 + (-i)^r*H3 (scale deferred)
        float Tre[4][3], T1im[4];
        #pragma unroll
        for (int fw = 0; fw < 3; ++fw) {
            float h0r = Hre[0][fw], h0i = Him[0][fw];
            float h1r = Hre[1][fw], h1i = Him[1][fw];
            float h2r = Hre[2][fw], h2i = Him[2][fw];
            float h3r = Hre[3][fw], h3i = Him[3][fw];
            Tre[0][fw] = h0r + h1r + h2r + h3r;
            Tre[1][fw] = h0r - h1i - h2r + h3i;
            Tre[2][fw] = h0r - h1r + h2r - h3r;
            Tre[3][fw] = h0r + h1i - h2r - h3i;
            if (fw == 1) {
                T1im[0] = h0i + h1i + h2i + h3i;
                T1im[1] = h0i + h1r - h2i - h3r;
                T1im[2] = h0i - h1i + h2i - h3i;
                T1im[3] = h0i - h1r - h2i + h3r;
            }
        }
        // irfft along width (C2R: imag of DC/Nyquist ignored), total scale 1/16
        #pragma unroll
        for (int r = 0; r < 4; ++r) {
            float t0 = Tre[r][0], t2 = Tre[r][2];
            float t1r = Tre[r][1], t1i = T1im[r];
            int gh = h0 + r;
            size_t obase = ((size_t)b * DIM + c) * IMG_H * IMG_W + (size_t)gh * IMG_W + w0 + bw * 4;
            int sbase = c * CST + (r + 1) * SROW + 4 + bw * 4;
            float4 o;
            o.x = (t0 + t2 + 2.f * t1r) * (1.f / 16.f) + xs[sbase + 0];
            o.y = (t0 - t2 - 2.f * t1i) * (1.f / 16.f) + xs[sbase + 1];
            o.z = (t0 + t2 - 2.f * t1r) * (1.f / 16.f) + xs[sbase + 2];
            o.w = (t0 - t2 + 2.f * t1i) * (1.f / 16.f) + xs[sbase + 3];
            *(float4*)(out + obase) = o;                 // 16B-aligned b128 store
        }
    }
}

extern "C" void kernel_launch(void* const* d_in, const int* in_sizes, int n_in,
                              void* d_out, int out_size, void* d_ws, size_t ws_size,
                              hipStream_t stream) {
    (void)in_sizes; (void)n_in; (void)out_size; (void)d_ws; (void)ws_size;
    const float* x      = (const float*)d_in[0];
    const float* w_in1  = (const float*)d_in[1];
    const float* w_dw   = (const float*)d_in[2];
    const float* w_in2  = (const float*)d_in[3];
    const float* w_freq = (const float*)d_in[4];
    float* out = (float*)d_out;

    dim3 grid(IMG_W / TW, IMG_H / TH, NBATCH);   // (8, 128, 4) = 4096 WGs
    edfdffn_fused<<<grid, NTHREADS, 0, stream>>>(x, w_in1, w_dw, w_in2, w_freq, out);
}